// RetentiveCrossEntropyLoss_90640989814992
// MI455X (gfx1250) — compile-verified
//
#include <hip/hip_runtime.h>
#include <math.h>

typedef float v4f __attribute__((ext_vector_type(4)));

namespace {

constexpr float kLog2e = 1.4426950408889634f;   // 1/ln(2)
constexpr float kLn2   = 0.6931471805599453f;

__device__ __forceinline__ float fast_exp2(float x) {
    return __builtin_amdgcn_exp2f(x);   // v_exp_f32: 2^x
}
__device__ __forceinline__ float fast_log2(float x) {
    return __builtin_amdgcn_logf(x);    // v_log_f32: log2(x)
}

// Online logsumexp update in log2 domain. v is pre-scaled by log2(e).
// Costs exactly one v_exp_f32 per element:
//   if v <= m : s = s + 2^(v-m)
//   else      : s = s*2^(m-v) + 1 ; m = v
__device__ __forceinline__ void online_update(float& m, float& s, float v) {
    float mx = fmaxf(m, v);
    float mn = fminf(m, v);
    float e  = fast_exp2(mn - mx);        // 2^(min-max), == 0 when m starts at -inf
    bool  keep = (v <= m);
    float mul = keep ? 1.0f : e;
    float add = keep ? e : 1.0f;
    s = __fmaf_rn(s, mul, add);
    m = mx;
}

// Merge two (max, sumexp) partials (log2 domain).
__device__ __forceinline__ void merge_pair(float& m, float& s, float mo, float so) {
    float mx = fmaxf(m, mo);
    float g  = (mx == -INFINITY) ? 0.0f : mx;  // avoid (-inf)-(-inf)=NaN for empty partials
    s = s * fast_exp2(m - g) + so * fast_exp2(mo - g);
    m = mx;
}

__device__ __forceinline__ void scalar_elem(const float* __restrict__ tp,
                                            const float* __restrict__ np,
                                            int j, int label,
                                            float& m1, float& s1,
                                            float& m2, float& s2) {
    float t = tp[j];
    float n = np[j];
    float h1 = (j == label) ? n : t;   // head1: target row, label slot <- new
    float h2 = (j == label) ? t : n;   // head2: new row, label slot <- target
    online_update(m1, s1, h1 * kLog2e);
    online_update(m2, s2, h2 * kLog2e);
}

__global__ __launch_bounds__(256)
void rce_two_head_kernel(const float* __restrict__ newl,
                         const float* __restrict__ tgt,
                         const int*   __restrict__ lab,
                         float*       __restrict__ out,
                         int V) {
    const int row  = blockIdx.x;
    const int tid  = threadIdx.x;
    const int nthr = blockDim.x;

    const size_t base = (size_t)row * (size_t)V;
    const int label   = lab[row];

    const float* __restrict__ tp = tgt  + base;
    const float* __restrict__ np = newl + base;

    // Two independent accumulator chains per head to hide v_exp_f32 latency.
    float m1a = -INFINITY, s1a = 0.0f, m1b = -INFINITY, s1b = 0.0f;
    float m2a = -INFINITY, s2a = 0.0f, m2b = -INFINITY, s2b = 0.0f;

    // ---- scalar prologue: reach 16B alignment (row base is only 4B aligned) ----
    int a = (int)((4u - ((unsigned)(base & 3u))) & 3u);
    if (a > V) a = V;
    if (tid < a) {
        scalar_elem(tp, np, tid, label, m1a, s1a, m2a, s2a);
    }

    // ---- main b128 streaming loop (nontemporal: zero reuse, don't pollute L2) ----
    const int nvec = (V - a) >> 2;
    const v4f* __restrict__ t4p = (const v4f*)(tp + a);
    const v4f* __restrict__ n4p = (const v4f*)(np + a);

    for (int k = tid; k < nvec; k += nthr) {
        // keep the HBM pipe primed one block-stride ahead -> global_prefetch_b8
        __builtin_prefetch(t4p + k + nthr, 0, 0);
        __builtin_prefetch(n4p + k + nthr, 0, 0);

        v4f t4 = __builtin_nontemporal_load(t4p + k);   // global_load_b128 th:NT
        v4f n4 = __builtin_nontemporal_load(n4p + k);

        const int j0 = a + 4 * k;
        unsigned d = (unsigned)(label - j0);
        if (d < 4u) {   // rare: exactly one lane per row per pass; swap t<->n at slot d
            switch (d) {
                case 0u: { float x = t4.x; t4.x = n4.x; n4.x = x; } break;
                case 1u: { float x = t4.y; t4.y = n4.y; n4.y = x; } break;
                case 2u: { float x = t4.z; t4.z = n4.z; n4.z = x; } break;
                default: { float x = t4.w; t4.w = n4.w; n4.w = x; } break;
            }
        }

        // head1 consumes (patched) target row, head2 consumes (patched) new row.
        online_update(m1a, s1a, t4.x * kLog2e);
        online_update(m1b, s1b, t4.y * kLog2e);
        online_update(m1a, s1a, t4.z * kLog2e);
        online_update(m1b, s1b, t4.w * kLog2e);

        online_update(m2a, s2a, n4.x * kLog2e);
        online_update(m2b, s2b, n4.y * kLog2e);
        online_update(m2a, s2a, n4.z * kLog2e);
        online_update(m2b, s2b, n4.w * kLog2e);
    }

    // ---- scalar tail ((V - a) % 4 leftovers) ----
    {
        int j = a + 4 * nvec + tid;
        if (j < V) {
            scalar_elem(tp, np, j, label, m1a, s1a, m2a, s2a);
        }
    }

    // fold the dual chains
    merge_pair(m1a, s1a, m1b, s1b);
    merge_pair(m2a, s2a, m2b, s2b);

    // ---- wave32 butterfly reduction (ds lane permutes) ----
    #pragma unroll
    for (int off = 16; off > 0; off >>= 1) {
        float om1 = __shfl_xor(m1a, off, 32);
        float os1 = __shfl_xor(s1a, off, 32);
        float om2 = __shfl_xor(m2a, off, 32);
        float os2 = __shfl_xor(s2a, off, 32);
        merge_pair(m1a, s1a, om1, os1);
        merge_pair(m2a, s2a, om2, os2);
    }

    // ---- cross-wave reduction via LDS (8 waves / block) ----
    __shared__ float sm1[16], ss1[16], sm2[16], ss2[16];
    const int wid = tid >> 5;
    if ((tid & 31) == 0) {
        sm1[wid] = m1a; ss1[wid] = s1a;
        sm2[wid] = m2a; ss2[wid] = s2a;
    }
    __syncthreads();

    if (tid == 0) {
        float M1 = sm1[0], S1 = ss1[0], M2 = sm2[0], S2 = ss2[0];
        const int nwaves = (nthr + 31) >> 5;
        for (int i = 1; i < nwaves; ++i) {
            merge_pair(M1, S1, sm1[i], ss1[i]);
            merge_pair(M2, S2, sm2[i], ss2[i]);
        }
        const float nl = np[label];
        const float tl = tp[label];
        const float lse1 = (M1 + fast_log2(S1)) * kLn2;
        const float lse2 = (M2 + fast_log2(S2)) * kLn2;
        out[row] = (lse1 - nl) + (lse2 - tl);
    }
}

} // namespace

extern "C" void kernel_launch(void* const* d_in, const int* in_sizes, int n_in,
                              void* d_out, int out_size, void* d_ws, size_t ws_size,
                              hipStream_t stream) {
    const float* newl = (const float*)d_in[0];   // new_logits   [B,V] f32
    const float* tgt  = (const float*)d_in[1];   // target_logits[B,V] f32
    const int*   lab  = (const int*)d_in[2];     // label        [B]   int

    float* out = (float*)d_out;                  // loss [B] f32

    const int B = in_sizes[2];
    const int V = (int)(in_sizes[0] / (B > 0 ? B : 1));

    rce_two_head_kernel<<<B, 256, 0, stream>>>(newl, tgt, lab, out, V);
}